// GPU_NTM_31525059953334
// MI455X (gfx1250) — compile-verified
//
#include <hip/hip_runtime.h>
#include <hip/hip_bf16.h>
#include <math.h>

// ---------------- problem dims ----------------
#define BB   128
#define TT   32
#define DIN  1024      // D1*D2
#define HH   1024
#define OO   512
#define RR   1
#define KK_  8
#define NN_  128
#define MM_  64
#define KN   1024      // KK_*NN_
#define KNM  65536     // KK_*NN_*MM_
#define EPSF 1e-8f

typedef float v2f __attribute__((ext_vector_type(2)));
typedef float v8f __attribute__((ext_vector_type(8)));

__device__ __forceinline__ float sigmoidf_(float x) { return 1.0f / (1.0f + expf(-x)); }
__device__ __forceinline__ float softplusf_(float x) {
  return (x > 20.0f) ? x : log1pf(expf(x));
}

// =====================================================================
// WMMA f32 GEMM:  Y = epilogue( X(128 x K) @ W(N x K)^T + bias )
// MODE 0: Y = acc + bias
// MODE 1: Y = relu(acc + bias)
// MODE 2: Y = relu(aux + relu(acc + bias))
// Block = 128 threads (4 waves); each wave owns one 16-row tile, all
// waves share the same 64-col stripe (NT 16x16 accumulators, NT<=4).
// Software-pipelined: ping-pong fragment groups of 16 K-steps so the
// next group's global_load_b64s are in flight during this group's WMMAs.
// =====================================================================
template <int MODE, int NT>
__global__ __launch_bounds__(128) void wmma_gemm_kernel(
    const float* __restrict__ X, int ldx,
    const float* __restrict__ W, int ldw,
    const float* __restrict__ bias,
    float* __restrict__ Y, int ldy,
    const float* __restrict__ aux, int ldaux,
    int n_base, int N, int K) {
  const int lane = threadIdx.x & 31;
  const int wave = threadIdx.x >> 5;
  const int m0   = (blockIdx.y * 4 + wave) << 4;    // row tile base
  const int n0   = n_base + blockIdx.x * 64;        // col stripe base

  const int lm  = lane & 15;                        // row-in-tile (A) / col (B,D)
  const int lkh = (lane >> 4) << 1;                 // K sub-offset: 0 or 2

  const float* xrow = X + (size_t)(m0 + lm) * ldx + lkh;
  const float* wrow[NT];
#pragma unroll
  for (int ct = 0; ct < NT; ++ct) {
    int jc = n0 + ct * 16 + lm;
    if (jc > N - 1) jc = N - 1;                     // clamp (stores masked)
    wrow[ct] = W + (size_t)jc * ldw + lkh;
  }

  v8f acc[NT];
#pragma unroll
  for (int ct = 0; ct < NT; ++ct) acc[ct] = (v8f){0.f,0.f,0.f,0.f,0.f,0.f,0.f,0.f};

  // fragment group = 4 k-steps of 4 (16 K values)
  v2f a0[4], a1[4], b0[4][NT], b1[4][NT];

#define LOADG(Af, Bf, OFF)                                        \
  _Pragma("unroll") for (int u = 0; u < 4; ++u) {                 \
    Af[u] = *(const v2f*)(xrow + (OFF) + 4 * u);                  \
    _Pragma("unroll") for (int ct = 0; ct < NT; ++ct)             \
      Bf[u][ct] = *(const v2f*)(wrow[ct] + (OFF) + 4 * u);        \
  }

#define WMMAG(Af, Bf)                                             \
  _Pragma("unroll") for (int u = 0; u < 4; ++u)                   \
    _Pragma("unroll") for (int ct = 0; ct < NT; ++ct)             \
      acc[ct] = __builtin_amdgcn_wmma_f32_16x16x4_f32(            \
          false, Af[u], false, Bf[u][ct], (short)0, acc[ct], false, false);

  LOADG(a0, b0, 0);
  for (int kb = 0; kb + 32 <= K; kb += 32) {
#pragma unroll
    for (int ct = 0; ct < NT; ++ct)
      __builtin_prefetch((const void*)(wrow[ct] + kb + 128), 0, 3);
    LOADG(a1, b1, kb + 16);                // in flight during group-0 WMMAs
    WMMAG(a0, b0);
    if (kb + 32 < K) { LOADG(a0, b0, kb + 32); }  // in flight during group-1 WMMAs
    WMMAG(a1, b1);
  }
#undef LOADG
#undef WMMAG

  const int rbase = (lane >> 4) << 3;               // 0 or 8
#pragma unroll
  for (int ct = 0; ct < NT; ++ct) {
    const int jc = n0 + ct * 16 + lm;
    if (jc >= N) continue;
    const float bv = bias ? bias[jc] : 0.0f;
#pragma unroll
    for (int r = 0; r < 8; ++r) {
      const int row = m0 + rbase + r;
      float v = acc[ct][r] + bv;
      if (MODE >= 1) v = fmaxf(v, 0.0f);
      if (MODE == 2) v = fmaxf(aux[(size_t)row * ldaux + jc] + v, 0.0f);
      Y[(size_t)row * ldy + jc] = v;
    }
  }
}

// =====================================================================
// Fused NTM step: head activations, read+write addressing, read vector,
// in-place memory update. One workgroup (256 threads) per batch element.
// =====================================================================
__device__ __forceinline__ float block_reduce(float v, float* s_red, int t, int domax) {
  s_red[t] = v;
  __syncthreads();
  for (int off = 128; off > 0; off >>= 1) {
    if (t < off) s_red[t] = domax ? fmaxf(s_red[t], s_red[t + off]) : (s_red[t] + s_red[t + off]);
    __syncthreads();
  }
  float r = s_red[0];
  __syncthreads();
  return r;
}

__global__ __launch_bounds__(256) void ntm_step_kernel(
    const float* __restrict__ rh,       // (B,70)
    const float* __restrict__ wh,       // (B,198)
    const float* __restrict__ wr_prev,  // (B,1024)
    const float* __restrict__ ww_prev,  // (B,1024)
    const float* __restrict__ m_prev,   // (B,65536)
    float* __restrict__ wr_out,
    float* __restrict__ ww_out,
    float* __restrict__ m_out,          // may alias m_prev (pointwise update)
    float* __restrict__ read_out)       // (B,64)
{
  const int b = blockIdx.x;
  const int t = threadIdx.x;

  __shared__ float s_kr[64], s_kw[64], s_er[64], s_ad[64];
  __shared__ float s_wA[KN];   // read-head weights (logits -> final)
  __shared__ float s_wB[KN];   // write-head weights
  __shared__ float s_tmp[KN];  // shift scratch
  __shared__ float s_red[256];
  __shared__ float s_scal[16];

  const float* rhb = rh + (size_t)b * 70;
  const float* whb = wh + (size_t)b * 198;

  if (t < 64)        s_kr[t]       = tanhf(rhb[t]);
  else if (t < 128)  s_kw[t - 64]  = tanhf(whb[t - 64]);
  else if (t < 192)  s_er[t - 128] = sigmoidf_(whb[70 + (t - 128)]);
  else               s_ad[t - 192] = tanhf(whb[134 + (t - 192)]);

  if (t == 0) {
    s_scal[0] = softplusf_(rhb[64]);
    s_scal[1] = sigmoidf_(rhb[65]);
    float a0 = rhb[66], a1 = rhb[67], a2 = rhb[68];
    float mx = fmaxf(a0, fmaxf(a1, a2));
    float e0 = expf(a0 - mx), e1 = expf(a1 - mx), e2 = expf(a2 - mx);
    float s = e0 + e1 + e2;
    s_scal[2] = e0 / s; s_scal[3] = e1 / s; s_scal[4] = e2 / s;
    s_scal[5] = 1.0f + softplusf_(rhb[69]);
  } else if (t == 1) {
    s_scal[6] = softplusf_(whb[64]);
    s_scal[7] = sigmoidf_(whb[65]);
    float a0 = whb[66], a1 = whb[67], a2 = whb[68];
    float mx = fmaxf(a0, fmaxf(a1, a2));
    float e0 = expf(a0 - mx), e1 = expf(a1 - mx), e2 = expf(a2 - mx);
    float s = e0 + e1 + e2;
    s_scal[8] = e0 / s; s_scal[9] = e1 / s; s_scal[10] = e2 / s;
    s_scal[11] = 1.0f + softplusf_(whb[69]);
  }
  __syncthreads();
  if (t == 0) { float s = 0.f; for (int j = 0; j < 64; ++j) s += s_kr[j] * s_kr[j]; s_scal[12] = sqrtf(s); }
  if (t == 1) { float s = 0.f; for (int j = 0; j < 64; ++j) s += s_kw[j] * s_kw[j]; s_scal[13] = sqrtf(s); }
  __syncthreads();

  // ---- Pass A: cosine-similarity logits for both heads (single m sweep) ----
  const float* mb = m_prev + (size_t)b * KNM;
  const float invKr = 1.0f / (s_scal[12] + EPSF);
  const float invKw = 1.0f / (s_scal[13] + EPSF);
  const float beta_r = s_scal[0], beta_w = s_scal[6];
#pragma unroll
  for (int q = 0; q < 4; ++q) {
    const int p = t + q * 256;
    const float* mv = mb + (size_t)p * MM_;
    float dr = 0.f, dw = 0.f, nn = 0.f;
#pragma unroll 8
    for (int j = 0; j < MM_; ++j) {
      float v = mv[j];
      dr += s_kr[j] * v;
      dw += s_kw[j] * v;
      nn += v * v;
    }
    const float inm = 1.0f / (sqrtf(nn) + EPSF);
    s_wA[p] = beta_r * dr * invKr * inm;
    s_wB[p] = beta_w * dw * invKw * inm;
  }
  __syncthreads();

  // ---- softmax / interpolate / shift / sharpen for each head ----
  for (int head = 0; head < 2; ++head) {
    float* s_w = head ? s_wB : s_wA;
    const float* prev = head ? (ww_prev + (size_t)b * KN) : (wr_prev + (size_t)b * KN);
    float* outp = head ? (ww_out + (size_t)b * KN) : (wr_out + (size_t)b * KN);
    const int o = head ? 6 : 0;
    const float g = s_scal[o + 1], s0 = s_scal[o + 2], s1 = s_scal[o + 3], s2 = s_scal[o + 4];
    const float gamma = s_scal[o + 5];

    float lv[4], mymax = -1e30f;
#pragma unroll
    for (int q = 0; q < 4; ++q) { lv[q] = s_w[t + q * 256]; mymax = fmaxf(mymax, lv[q]); }
    const float mx = block_reduce(mymax, s_red, t, 1);
    float psum = 0.f;
#pragma unroll
    for (int q = 0; q < 4; ++q) { lv[q] = expf(lv[q] - mx); psum += lv[q]; }
    const float inv_sum = 1.0f / block_reduce(psum, s_red, t, 0);
#pragma unroll
    for (int q = 0; q < 4; ++q) {
      const int p = t + q * 256;
      s_tmp[p] = g * lv[q] * inv_sum + (1.0f - g) * prev[p];   // wg
    }
    __syncthreads();
    float wpv[4], wsum = 0.f;
#pragma unroll
    for (int q = 0; q < 4; ++q) {
      const int p = t + q * 256;
      const int n = p & (NN_ - 1);
      const int kb = p - n;
      const float wsh = s0 * s_tmp[kb + ((n + 1) & (NN_ - 1))]
                      + s1 * s_tmp[p]
                      + s2 * s_tmp[kb + ((n - 1) & (NN_ - 1))];
      wpv[q] = powf(wsh + EPSF, gamma);
      wsum += wpv[q];
    }
    const float inv_tot = 1.0f / block_reduce(wsum, s_red, t, 0);
#pragma unroll
    for (int q = 0; q < 4; ++q) {
      const int p = t + q * 256;
      const float f = wpv[q] * inv_tot;
      s_w[p] = f;
      outp[p] = f;
    }
    __syncthreads();
  }

  // ---- Pass B: read vector  read[j] = sum_p wr[p] * m[p, j] ----
  {
    const int j = t & 63;
    const int grp = t >> 6;
    float partial = 0.f;
    const int p0 = grp * 256;
    for (int p = p0; p < p0 + 256; ++p) partial += s_wA[p] * mb[(size_t)p * MM_ + j];
    s_red[t] = partial;
    __syncthreads();
    if (t < 64)
      read_out[(size_t)b * MM_ + t] = s_red[t] + s_red[64 + t] + s_red[128 + t] + s_red[192 + t];
    __syncthreads();
  }

  // ---- memory update (pointwise; safe in-place) ----
  float* mo = m_out + (size_t)b * KNM;
  for (int i = t; i < KNM; i += 256) {
    const int kn = i >> 6;
    const int j = i & 63;
    const float v = mb[i];
    const float w = s_wB[kn];
    mo[i] = v * (1.0f - w * s_er[j]) + w * s_ad[j];
  }
}

// =====================================================================
// GRU gate combine
// =====================================================================
__global__ __launch_bounds__(256) void gru_combine_kernel(
    const float* __restrict__ gi, const float* __restrict__ gh,
    const float* __restrict__ h_in, float* __restrict__ h_out) {
  const int idx = blockIdx.x * 256 + threadIdx.x;   // 0 .. B*H-1
  const int b = idx >> 10;
  const int i = idx & 1023;
  const float* gib = gi + (size_t)b * 3072;
  const float* ghb = gh + (size_t)b * 3072;
  const float ir = gib[i], iz = gib[i + HH], inn = gib[i + 2 * HH];
  const float hr = ghb[i], hz = ghb[i + HH], hn = ghb[i + 2 * HH];
  const float rg = sigmoidf_(ir + hr);
  const float z  = sigmoidf_(iz + hz);
  const float ng = tanhf(inn + rg * hn);
  h_out[idx] = (1.0f - z) * ng + z * h_in[idx];
}

// =====================================================================
// host-side GEMM dispatch: full 64-col stripes (NT=4) + tail (NT=1..3)
// =====================================================================
template <int MODE>
static inline void launch_gemm(const float* X, int ldx, const float* W, int ldw,
                               const float* bias, float* Y, int ldy,
                               const float* aux, int ldaux, int N, int K,
                               hipStream_t stream) {
  const dim3 blk(128);
  const int nfull = N >> 6;                     // full 64-col stripes
  if (nfull > 0)
    wmma_gemm_kernel<MODE, 4><<<dim3(nfull, 2), blk, 0, stream>>>(
        X, ldx, W, ldw, bias, Y, ldy, aux, ldaux, 0, N, K);
  const int rem = N - (nfull << 6);
  if (rem > 0) {
    const int nb = nfull << 6;
    const int nt = (rem + 15) >> 4;
    const dim3 g(1, 2);
    if (nt == 1)
      wmma_gemm_kernel<MODE, 1><<<g, blk, 0, stream>>>(X, ldx, W, ldw, bias, Y, ldy, aux, ldaux, nb, N, K);
    else if (nt == 2)
      wmma_gemm_kernel<MODE, 2><<<g, blk, 0, stream>>>(X, ldx, W, ldw, bias, Y, ldy, aux, ldaux, nb, N, K);
    else
      wmma_gemm_kernel<MODE, 3><<<g, blk, 0, stream>>>(X, ldx, W, ldw, bias, Y, ldy, aux, ldaux, nb, N, K);
  }
}

// =====================================================================
// host launcher
// =====================================================================
extern "C" void kernel_launch(void* const* d_in, const int* in_sizes, int n_in,
                              void* d_out, int out_size, void* d_ws, size_t ws_size,
                              hipStream_t stream) {
  (void)in_sizes; (void)n_in; (void)out_size; (void)ws_size;
  const float* x        = (const float*)d_in[0];
  const float* h0       = (const float*)d_in[1];
  const float* wr0      = (const float*)d_in[2];
  const float* ww0      = (const float*)d_in[3];
  const float* m0       = (const float*)d_in[4];
  const float* W_rt     = (const float*)d_in[5];
  const float* b_rt     = (const float*)d_in[6];
  const float* W_ih     = (const float*)d_in[7];
  const float* W_hh     = (const float*)d_in[8];
  const float* b_ih     = (const float*)d_in[9];
  const float* b_hh     = (const float*)d_in[10];
  const float* W_rhead  = (const float*)d_in[11];
  const float* b_rhead  = (const float*)d_in[12];
  const float* W_whead  = (const float*)d_in[13];
  const float* b_whead  = (const float*)d_in[14];
  const float* W_out    = (const float*)d_in[15];
  const float* b_out    = (const float*)d_in[16];

  // d_out tuple regions: outs(B,T,O), h(B,H), wr(B,1,K,N), ww(B,K,N), m(B,K,N,M)
  float* o_outs = (float*)d_out;
  float* o_h    = o_outs + (size_t)BB * TT * OO;
  float* o_wr   = o_h    + (size_t)BB * HH;
  float* o_ww   = o_wr   + (size_t)BB * KN;
  float* o_m    = o_ww   + (size_t)BB * KN;

  // workspace layout
  float* w      = (float*)d_ws;
  float* ws_rh  = w;                 w += (size_t)BB * 70;
  float* ws_wh  = w;                 w += (size_t)BB * 198;
  float* ws_gh  = w;                 w += (size_t)BB * 3 * HH;
  float* ws_gi  = w;                 w += (size_t)BB * 3 * HH;
  float* ws_xc  = w;                 w += (size_t)BB * DIN;
  float* ws_rd  = w;                 w += (size_t)BB * MM_;

  for (int t = 0; t < TT; ++t) {
    const float* hcur = (t == 0) ? h0 : o_h;

    // h-dependent projections
    launch_gemm<0>(hcur, HH, W_rhead, HH, b_rhead, ws_rh, 70, nullptr, 0, 70, HH, stream);
    launch_gemm<0>(hcur, HH, W_whead, HH, b_whead, ws_wh, 198, nullptr, 0, 198, HH, stream);
    launch_gemm<0>(hcur, HH, W_hh, HH, b_hh, ws_gh, 3 * HH, nullptr, 0, 3 * HH, HH, stream);

    // fused NTM addressing + read + memory update
    ntm_step_kernel<<<BB, 256, 0, stream>>>(
        ws_rh, ws_wh,
        (t == 0) ? wr0 : o_wr, (t == 0) ? ww0 : o_ww, (t == 0) ? m0 : o_m,
        o_wr, o_ww, o_m, ws_rd);

    // xc = relu(x_t + relu(read @ W_rt^T + b_rt))
    launch_gemm<2>(ws_rd, MM_, W_rt, MM_, b_rt, ws_xc, DIN,
                   x + (size_t)t * DIN, TT * DIN, DIN, MM_, stream);

    // gi = xc @ W_ih^T + b_ih
    launch_gemm<0>(ws_xc, DIN, W_ih, DIN, b_ih, ws_gi, 3 * HH, nullptr, 0, 3 * HH, DIN, stream);

    // GRU combine -> new h (in d_out region)
    gru_combine_kernel<<<(BB * HH) / 256, 256, 0, stream>>>(ws_gi, ws_gh, hcur, o_h);

    // out_t = relu(h_n @ W_out^T + b_out), written into (B,T,O) layout
    launch_gemm<1>(o_h, HH, W_out, HH, b_out, o_outs + (size_t)t * OO, TT * OO,
                   nullptr, 0, OO, HH, stream);
  }
}